// ScaledDotProductAttention_6330781794585
// MI455X (gfx1250) — compile-verified
//
#include <hip/hip_runtime.h>
#include <hip/hip_bf16.h>

// ---------------------------------------------------------------------------
// Scaled dot-product attention, fp32, returning (output, attention).
//   B=4 H=16 S=2048 D=64, scale = 0.125
//
// Roofline: must write the full attention matrix (1.07 GB) -> memory bound
// (~50us at 23.3 TB/s). Strategy: one WG per (b,h,16-row query tile); full
// 16x2048 score block lives in LDS (131 KB of 320 KB/WGP) so attention is
// written to HBM exactly once. Matmuls use V_WMMA_F32_16X16X4_F32 (exact
// fp32, matches reference precision; compute is only ~69 GFLOP total).
// ---------------------------------------------------------------------------

#define Bsz 4
#define Hn 16
#define Sn 2048
#define Dn 64
#define ATTN_SCALE 0.125f

#define QT 16                      // query rows per workgroup
#define NWAVES 4
#define NTHREADS (NWAVES * 32)     // wave32
#define KCOLS_PER_WAVE (Sn / NWAVES)   // 512 key columns per wave
#define SPAD 2052                  // LDS row stride (2048+4): 16 rows -> 16 distinct banks

typedef __attribute__((ext_vector_type(2))) float v2f;
typedef __attribute__((ext_vector_type(8))) float v8f;

__global__ __launch_bounds__(NTHREADS)
void ScaledDotProductAttention_6330781794585_kernel(
    const float* __restrict__ q, const float* __restrict__ k,
    const float* __restrict__ v, float* __restrict__ out,
    float* __restrict__ attn)
{
    __shared__ float sc[QT * SPAD];     // 16 x 2048 score block (padded)
    __shared__ float redmax[QT][8];
    __shared__ float redsum[QT][8];
    __shared__ float invrow[QT];

    const int tid  = threadIdx.x;
    const int wave = tid >> 5;
    const int lane = tid & 31;
    const int lm   = lane & 15;    // A row (or B/C column) within tile
    const int kh   = lane >> 4;    // K-half selector per WMMA A/B layout

    const int qtiles = Sn / QT;                 // 128
    const int qt_id  = blockIdx.x % qtiles;
    const int bh     = blockIdx.x / qtiles;
    const int qrow0  = qt_id * QT;

    const float* qbase = q + ((size_t)bh * Sn + qrow0) * Dn;
    const float* kbase = k + (size_t)bh * Sn * Dn;
    const float* vbase = v + (size_t)bh * Sn * Dn;
    float* outbase  = out  + ((size_t)bh * Sn + qrow0) * Dn;
    float* attnbase = attn + ((size_t)bh * Sn + qrow0) * Sn;

    // ---- Load Q A-fragments once; reused across all 128 key tiles.
    // A 16x4 f32 layout: VGPR v, lane-half h holds K = v + 2h.
    v2f aq[16];
#pragma unroll
    for (int t = 0; t < 16; ++t) {
        aq[t] = *(const v2f*)(qbase + lm * Dn + 4 * t + 2 * kh);
    }

    // ---- Phase 1: scores = (Q K^T) * scale, each wave owns 512 key columns.
    for (int kt = 0; kt < KCOLS_PER_WAVE / 16; ++kt) {
        const int col0 = wave * KCOLS_PER_WAVE + kt * 16;
        v8f c = {};
#pragma unroll
        for (int t = 0; t < 16; ++t) {
            // B 4x16 fragment of K^T: element (kd, n) = K[col0+n][kd]
            v2f b = *(const v2f*)(kbase + (size_t)(col0 + lm) * Dn + 4 * t + 2 * kh);
            c = __builtin_amdgcn_wmma_f32_16x16x4_f32(
                    false, aq[t], false, b, (short)0, c, false, false);
        }
#pragma unroll
        for (int i = 0; i < 8; ++i) {
            const int row = i + 8 * kh;      // C layout: VGPR i -> M = i + 8h
            sc[row * SPAD + col0 + lm] = c[i] * ATTN_SCALE;
        }
    }
    __syncthreads();

    // ---- Phase 2: softmax over 2048 columns. 8 threads per row.
    const int srow = tid & 15;
    const int part = tid >> 4;               // 0..7
    const int c0   = part * (Sn / 8);        // 256-column slice
    float mx = -3.402823466e38f;
    for (int c = 0; c < Sn / 8; ++c)
        mx = fmaxf(mx, sc[srow * SPAD + c0 + c]);
    redmax[srow][part] = mx;
    __syncthreads();
    mx = redmax[srow][0];
#pragma unroll
    for (int p = 1; p < 8; ++p) mx = fmaxf(mx, redmax[srow][p]);

    float sum = 0.0f;
    for (int c = 0; c < Sn / 8; ++c) {
        float e = __expf(sc[srow * SPAD + c0 + c] - mx);
        sc[srow * SPAD + c0 + c] = e;        // keep unnormalized P in LDS
        sum += e;
    }
    redsum[srow][part] = sum;
    __syncthreads();
    sum = 0.0f;
#pragma unroll
    for (int p = 0; p < 8; ++p) sum += redsum[srow][p];
    if (part == 0) invrow[srow] = 1.0f / sum;
    __syncthreads();

    // ---- Phase 3: write normalized attention (coalesced, written once).
#pragma unroll 4
    for (int i = 0; i < (QT * Sn) / NTHREADS; ++i) {
        const int idx = tid + i * NTHREADS;
        const int row = idx >> 11;           // / 2048
        const int col = idx & (Sn - 1);
        attnbase[idx] = sc[row * SPAD + col] * invrow[row];
    }

    // ---- Phase 4: O = P @ V (unnormalized P), row-scale at the end.
    // Each wave owns one 16-wide N tile of D=64.
    const int n0 = wave * 16;
    v8f co = {};
    for (int kb = 0; kb < Sn; kb += 4) {
        // A fragment from LDS (padded stride -> conflict-free across 16 rows)
        v2f a = *(const v2f*)&sc[lm * SPAD + kb + 2 * kh];
        v2f b;
        b.x = vbase[(size_t)(kb + 2 * kh) * Dn + n0 + lm];
        b.y = vbase[(size_t)(kb + 2 * kh + 1) * Dn + n0 + lm];
        co = __builtin_amdgcn_wmma_f32_16x16x4_f32(
                 false, a, false, b, (short)0, co, false, false);
    }
#pragma unroll
    for (int i = 0; i < 8; ++i) {
        const int row = i + 8 * kh;
        outbase[row * Dn + n0 + lm] = co[i] * invrow[row];
    }
}

extern "C" void kernel_launch(void* const* d_in, const int* in_sizes, int n_in,
                              void* d_out, int out_size, void* d_ws, size_t ws_size,
                              hipStream_t stream) {
    (void)in_sizes; (void)n_in; (void)d_ws; (void)ws_size; (void)out_size;
    const float* q = (const float*)d_in[0];
    const float* k = (const float*)d_in[1];
    const float* v = (const float*)d_in[2];
    float* out  = (float*)d_out;
    float* attn = out + (size_t)Bsz * Hn * Sn * Dn;   // outputs concatenated flat

    const int nblocks = Bsz * Hn * (Sn / QT);         // 8192
    ScaledDotProductAttention_6330781794585_kernel<<<nblocks, NTHREADS, 0, stream>>>(
        q, k, v, out, attn);
}